// QuantumCircuit_2602750182061
// MI455X (gfx1250) — compile-verified
//
#include <hip/hip_runtime.h>
#include <hip/hip_bf16.h>

typedef float v2f __attribute__((ext_vector_type(2)));
typedef float v8f __attribute__((ext_vector_type(8)));

// ---------------------------------------------------------------------------
// Setup kernel (runs once, 16 threads): build the fixed 16x16 complex unitary
// of the two StronglyEntanglingLayers (8 Rot + 8 CNOT), stored transposed
// (B = U^T):  B[k*16+n] = U[n][k] = (circuit applied to e_k)_n.
// Wire w occupies bit (3-w) of the flat index (wire 0 most significant).
// ---------------------------------------------------------------------------
__global__ void build_unitary_kernel(const float* __restrict__ p,
                                     float* __restrict__ Bre,
                                     float* __restrict__ Bim) {
  const int k = threadIdx.x;
  if (k >= 16) return;
  float sr[16], si[16];
#pragma unroll
  for (int i = 0; i < 16; ++i) { sr[i] = (i == k) ? 1.0f : 0.0f; si[i] = 0.0f; }
#pragma unroll
  for (int l = 0; l < 2; ++l) {
    // Rot(phi,theta,omega) = RZ(omega) RY(theta) RZ(phi) on each wire
#pragma unroll
    for (int w = 0; w < 4; ++w) {
      const float phi = p[l * 12 + w * 3 + 0];
      const float th  = p[l * 12 + w * 3 + 1];
      const float om  = p[l * 12 + w * 3 + 2];
      float c, s, ca, sa, cb, sb;
      __sincosf(0.5f * th, &s, &c);
      __sincosf(0.5f * (phi + om), &sa, &ca);
      __sincosf(0.5f * (phi - om), &sb, &cb);
      const float u00r =  c * ca, u00i = -c * sa;   // ep*c
      const float u01r = -s * cb, u01i = -s * sb;   // -conj(em)*s
      const float u10r =  s * cb, u10i = -s * sb;   // em*s
      const float u11r =  c * ca, u11i =  c * sa;   // conj(ep)*c
      const int mask = 8 >> w;
#pragma unroll
      for (int i = 0; i < 16; ++i) {
        if ((i & mask) == 0) {
          const int i1 = i | mask;
          const float a0r = sr[i],  a0i = si[i];
          const float a1r = sr[i1], a1i = si[i1];
          sr[i]  = u00r * a0r - u00i * a0i + u01r * a1r - u01i * a1i;
          si[i]  = u00r * a0i + u00i * a0r + u01r * a1i + u01i * a1r;
          sr[i1] = u10r * a0r - u10i * a0i + u11r * a1r - u11i * a1i;
          si[i1] = u10r * a0i + u10i * a0r + u11r * a1i + u11i * a1r;
        }
      }
    }
    // CNOT ring, range r = l+1
    const int r = (l == 0) ? 1 : 2;
#pragma unroll
    for (int w = 0; w < 4; ++w) {
      const int tgt   = (w + r) & 3;
      const int cmask = 8 >> w;
      const int tmask = 8 >> tgt;
#pragma unroll
      for (int i = 0; i < 16; ++i) {
        if ((i & cmask) != 0 && (i & tmask) == 0) {
          const int i1 = i | tmask;
          float tr = sr[i]; sr[i] = sr[i1]; sr[i1] = tr;
          float ti = si[i]; si[i] = si[i1]; si[i1] = ti;
        }
      }
    }
  }
#pragma unroll
  for (int n = 0; n < 16; ++n) {
    Bre[k * 16 + n] = sr[n];
    Bim[k * 16 + n] = si[n];
  }
}

// ---------------------------------------------------------------------------
// Main kernel: one wave per 16-element batch tile.
//   D[m,n] = sum_k psi0[m,k] * B[k,n],  B = U^T, complex via 4 real WMMAs/Kstep.
// A layout (ISA 16x4 f32): lane holds M=lane%16; this lane's K = 4s + 2*hi + j
// (j = .x/.y slot).  psi0 amplitude for index k:
//   amp(k) = (-i)^popcount(k) * prod_w (bit_w ? sin(x_w/2) : cos(x_w/2))
// with bit3..bit0 of k = wires 0..3; bit1 = hi, bit0 = j, bits3:2 = s.
// So: magnitude = p01[s] * (hi? s2:c2) * (j? s3:c3), phase = (-i)^(pop(s)+j+hi).
// Build the hi=0 amplitude at compile time, apply the extra (-i) for the hi
// half as (re,im) -> (im,-re): one v_cndmask per component, no arrays.
// ---------------------------------------------------------------------------
__global__ void __launch_bounds__(256)
qsim_wmma_kernel(const float* __restrict__ x,
                 const float* __restrict__ BreG,
                 const float* __restrict__ BimG,
                 float* __restrict__ out,
                 int nelem, int ntiles) {
  const int  lane = threadIdx.x & 31;
  const int  n    = lane & 15;
  const bool hi   = lane >= 16;
  const int  wid    = blockIdx.x * (blockDim.x >> 5) + (threadIdx.x >> 5);
  const int  nwaves = gridDim.x * (blockDim.x >> 5);

  // Load B = U^T operand registers once per wave (2 KB table, L1-resident).
  v2f Bre[4], Bim[4], BimN[4];
#pragma unroll
  for (int s4 = 0; s4 < 4; ++s4) {
    const int k0 = s4 * 4 + (hi ? 2 : 0);
    Bre[s4].x = BreG[(k0    ) * 16 + n];
    Bre[s4].y = BreG[(k0 + 1) * 16 + n];
    Bim[s4].x = BimG[(k0    ) * 16 + n];
    Bim[s4].y = BimG[(k0 + 1) * 16 + n];
    BimN[s4].x = -Bim[s4].x;   // F32 WMMA has no A/B NEG modifier: pre-negate
    BimN[s4].y = -Bim[s4].y;
  }
  // <Z0 Z1> sign for state index n: (-1)^(b0+b1), b0=bit3, b1=bit2
  const float zsign = (((n >> 3) ^ (n >> 2)) & 1) ? -1.0f : 1.0f;

  for (int tile = wid; tile < ntiles; tile += nwaves) {
    const int b  = tile * 16 + n;                 // batch element for this lane
    const int bc = (b < nelem) ? b : (nelem - 1); // clamp (keeps EXEC full)
    const float4 xv = reinterpret_cast<const float4*>(x)[bc];

    float c0, s0, c1, s1, c2, s2, c3, s3;
    __sincosf(0.5f * xv.x, &s0, &c0);
    __sincosf(0.5f * xv.y, &s1, &c1);
    __sincosf(0.5f * xv.z, &s2, &c2);
    __sincosf(0.5f * xv.w, &s3, &c3);

    // p01[s] for s = b0*2+b1 (bits 3:2 of k)
    const float p01_0 = c0 * c1, p01_1 = c0 * s1, p01_2 = s0 * c1, p01_3 = s0 * s1;
    // wire-2 factor depends on hi (bit1 of k): single select for the wave half
    const float f2  = hi ? s2 : c2;
    const float q23_0 = f2 * c3;   // j = 0 (bit0 = 0)
    const float q23_1 = f2 * s3;   // j = 1 (bit0 = 1)

    // Build A slot: qb = (popcount(s)+j) (phase with hi=0), magnitude mm.
    // hi=0: amp = mm*(-i)^qb ; hi=1: extra (-i): (re,im) -> (im,-re).
    auto mkA = [&](int qb, float mm, float* re, float* im) {
      float r0, i0, r1, i1;
      switch (qb & 3) {
        case 0:  r0 =  mm; i0 = 0.f; r1 = 0.f; i1 = -mm; break;
        case 1:  r0 = 0.f; i0 = -mm; r1 = -mm; i1 = 0.f; break;
        case 2:  r0 = -mm; i0 = 0.f; r1 = 0.f; i1 =  mm; break;
        default: r0 = 0.f; i0 =  mm; r1 =  mm; i1 = 0.f; break;
      }
      *re = hi ? r1 : r0;
      *im = hi ? i1 : i0;
    };

    v8f Dre = {};
    v8f Dim = {};
#pragma unroll
    for (int s4 = 0; s4 < 4; ++s4) {
      const float p01s = (s4 == 0) ? p01_0 : (s4 == 1) ? p01_1 : (s4 == 2) ? p01_2 : p01_3;
      const int   pops = (s4 & 1) + ((s4 >> 1) & 1);
      const float m0 = p01s * q23_0;   // k = 4*s4 + 2*hi + 0
      const float m1 = p01s * q23_1;   // k = 4*s4 + 2*hi + 1
      float arx, aix, ary, aiy;
      mkA(pops + 0, m0, &arx, &aix);
      mkA(pops + 1, m1, &ary, &aiy);
      v2f Are, Aim;
      Are.x = arx; Are.y = ary;
      Aim.x = aix; Aim.y = aiy;
      // Dre += Are*Bre - Aim*Bim ; Dim += Are*Bim + Aim*Bre
      Dre = __builtin_amdgcn_wmma_f32_16x16x4_f32(false, Are, false, Bre[s4],  (short)0, Dre, false, false);
      Dre = __builtin_amdgcn_wmma_f32_16x16x4_f32(false, Aim, false, BimN[s4], (short)0, Dre, false, false);
      Dim = __builtin_amdgcn_wmma_f32_16x16x4_f32(false, Are, false, Bim[s4],  (short)0, Dim, false, false);
      Dim = __builtin_amdgcn_wmma_f32_16x16x4_f32(false, Aim, false, Bre[s4],  (short)0, Dim, false, false);
    }

    // out[m] = sum_n zsign(n) * |D[m,n]|^2 ; reduce across each 16-lane half.
    // D VGPR j holds row m = 8*hi + j (N = n = lane%16).
#pragma unroll
    for (int j = 0; j < 8; ++j) {
      float t = zsign * (Dre[j] * Dre[j] + Dim[j] * Dim[j]);
      t += __shfl_xor(t, 1, 32);
      t += __shfl_xor(t, 2, 32);
      t += __shfl_xor(t, 4, 32);
      t += __shfl_xor(t, 8, 32);
      const int mrow = (hi ? 8 : 0) + j;
      const int ob   = tile * 16 + mrow;
      if (n == j && ob < nelem) out[ob] = t;
    }
  }
}

extern "C" void kernel_launch(void* const* d_in, const int* in_sizes, int n_in,
                              void* d_out, int out_size, void* d_ws, size_t ws_size,
                              hipStream_t stream) {
  const float* x      = (const float*)d_in[0];   // (B,4) float32
  const float* params = (const float*)d_in[1];   // (2,4,3) float32
  float* out = (float*)d_out;                    // (B,) float32
  const int nelem = in_sizes[0] / 4;

  float* Bre = (float*)d_ws;        // 256 floats
  float* Bim = Bre + 256;           // 256 floats  (2 KB total scratch)

  build_unitary_kernel<<<1, 32, 0, stream>>>(params, Bre, Bim);

  const int ntiles = (nelem + 15) / 16;
  int blocks = (ntiles + 7) / 8;    // 8 waves per block, 1 tile per wave
  if (blocks > 2048) blocks = 2048; // wave-stride loop covers the rest
  if (blocks < 1) blocks = 1;
  qsim_wmma_kernel<<<blocks, 256, 0, stream>>>(x, Bre, Bim, out, nelem, ntiles);
}